// HomogeneousMoELayer_20289425506413
// MI455X (gfx1250) — compile-verified
//
#include <hip/hip_runtime.h>
#include <math.h>

// ---------------- problem constants ----------------
#define T_TOKENS 8192          // B*S = 4*2048
#define DM       1024          // d_model
#define DH       512           // gating hidden
#define DF       4096          // d_ff
#define NUM_E    8
#define LN_EPS   1e-5f

// ---------------- expert-kernel tiling ----------------
#define MT   32                // token rows per block (two 16-row WMMA tiles)
#define FC   128               // F-chunk (per-iteration slice of d_ff)
#define SXS  1028              // padded LDS row stride for x tile (floats), 16B-multiple
#define HS   132               // padded LDS row stride for h chunk (floats)
#define EXPERT_SMEM_BYTES ((MT*SXS + MT*HS)*4 + MT*4 + MT*4)

typedef float v2f __attribute__((ext_vector_type(2)));
typedef float v8f __attribute__((ext_vector_type(8)));

// Branch-free erf-GELU (Abramowitz–Stegun 7.1.26, |err| <= 1.5e-7).
// One v_rcp_f32 + one v_exp_f32 + ~10 FMAs, no exec divergence in the hot loop.
__device__ __forceinline__ float gelu_erf(float x) {
    const float z  = x * 0.70710678118654752f;    // x / sqrt(2)
    const float az = fabsf(z);
    const float t  = __builtin_amdgcn_rcpf(fmaf(0.3275911f, az, 1.0f));
    float p = fmaf(t, 1.061405429f, -1.453152027f);
    p = fmaf(t, p, 1.421413741f);
    p = fmaf(t, p, -0.284496736f);
    p = fmaf(t, p, 0.254829592f);
    p *= t;
    const float er = copysignf(1.0f - p * __expf(-az * az), z);
    return 0.5f * x * (1.0f + er);
}

// CDNA5 async global->LDS copy (ASYNCcnt path), 16 bytes per lane.
// VDST operand carries the LDS byte offset = low 32 bits of the flat pointer
// (ISA 10.2).
__device__ __forceinline__ void async_g2l_b128(const float* g, float* l) {
    uint32_t loff = (uint32_t)(uintptr_t)l;
    uint64_t ga   = (uint64_t)(uintptr_t)g;
    asm volatile("global_load_async_to_lds_b128 %0, %1, off"
                 :: "v"(loff), "v"(ga) : "memory");
}
__device__ __forceinline__ void wait_asynccnt0() {
    asm volatile("s_wait_asynccnt 0" ::: "memory");
}

// ======================================================================
// 0) zero the routing counters
// ======================================================================
__global__ void moe_zero_kernel(int* __restrict__ counts) {
    if (threadIdx.x < NUM_E) counts[threadIdx.x] = 0;
}

// ======================================================================
// 1) gating: logits -> probs (for loss), top-2 gates, routing scatter
// ======================================================================
__global__ void moe_gating_kernel(const float* __restrict__ x,
                                  const float* __restrict__ gw1,
                                  const float* __restrict__ gb1,
                                  const float* __restrict__ gw2,
                                  const float* __restrict__ gb2,
                                  float* __restrict__ probs,     // [T,8]
                                  float* __restrict__ gates,     // [T,2]
                                  int*   __restrict__ counts,    // [8]
                                  int*   __restrict__ bucket)    // [8][T] holds t*2+slot
{
    const int t   = blockIdx.x;
    const int tid = threadIdx.x;
    __shared__ float sx[DM];
    __shared__ float sh[DH];
    __shared__ float slog[NUM_E];

    for (int i = tid; i < DM; i += 256) sx[i] = x[(size_t)t * DM + i];
    __syncthreads();

    for (int h = tid; h < DH; h += 256) {
        float acc = gb1[h];
        #pragma unroll 8
        for (int d = 0; d < DM; ++d) acc = fmaf(sx[d], gw1[(size_t)d * DH + h], acc);
        sh[h] = fmaxf(acc, 0.0f);
    }
    __syncthreads();

    if (tid < NUM_E) {
        float acc = gb2[tid];
        #pragma unroll 8
        for (int f = 0; f < DH; ++f) acc = fmaf(sh[f], gw2[(size_t)f * NUM_E + tid], acc);
        slog[tid] = acc;
    }
    __syncthreads();

    if (tid == 0) {
        float lg[NUM_E];
        #pragma unroll
        for (int e = 0; e < NUM_E; ++e) lg[e] = slog[e];

        float m = lg[0];
        #pragma unroll
        for (int e = 1; e < NUM_E; ++e) m = fmaxf(m, lg[e]);
        float s = 0.0f;
        float ex[NUM_E];
        #pragma unroll
        for (int e = 0; e < NUM_E; ++e) { ex[e] = __expf(lg[e] - m); s += ex[e]; }
        float inv = 1.0f / s;
        #pragma unroll
        for (int e = 0; e < NUM_E; ++e) probs[(size_t)t * NUM_E + e] = ex[e] * inv;

        // top-2 (stable: first index wins ties, like lax.top_k)
        int i0 = 0;
        #pragma unroll
        for (int e = 1; e < NUM_E; ++e) if (lg[e] > lg[i0]) i0 = e;
        int i1 = (i0 == 0) ? 1 : 0;
        #pragma unroll
        for (int e = 0; e < NUM_E; ++e) if (e != i0 && lg[e] > lg[i1]) i1 = e;

        float e1 = __expf(lg[i1] - lg[i0]);
        float g0 = 1.0f / (1.0f + e1);
        float g1 = e1 * g0;
        gates[(size_t)t * 2 + 0] = g0;
        gates[(size_t)t * 2 + 1] = g1;

        int p0 = atomicAdd(&counts[i0], 1);
        bucket[(size_t)i0 * T_TOKENS + p0] = t * 2 + 0;
        int p1 = atomicAdd(&counts[i1], 1);
        bucket[(size_t)i1 * T_TOKENS + p1] = t * 2 + 1;
    }
}

// ======================================================================
// 2) load-balancing loss: KL(uniform || mean probs), deterministic reduce
// ======================================================================
__global__ void moe_loss_kernel(const float* __restrict__ probs,
                                float* __restrict__ out_loss)
{
    const int tid = threadIdx.x;
    __shared__ float red[256 * NUM_E];
    float acc[NUM_E];
    #pragma unroll
    for (int e = 0; e < NUM_E; ++e) acc[e] = 0.0f;
    for (int t = tid; t < T_TOKENS; t += 256) {
        #pragma unroll
        for (int e = 0; e < NUM_E; ++e) acc[e] += probs[(size_t)t * NUM_E + e];
    }
    #pragma unroll
    for (int e = 0; e < NUM_E; ++e) red[tid * NUM_E + e] = acc[e];
    __syncthreads();
    for (int s = 128; s > 0; s >>= 1) {
        if (tid < s) {
            #pragma unroll
            for (int e = 0; e < NUM_E; ++e)
                red[tid * NUM_E + e] += red[(tid + s) * NUM_E + e];
        }
        __syncthreads();
    }
    if (tid == 0) {
        const float u = 1.0f / (float)NUM_E;
        const float lu = logf(u);
        float loss = 0.0f;
        #pragma unroll
        for (int e = 0; e < NUM_E; ++e) {
            float ep = red[e] / (float)T_TOKENS;
            loss += u * (lu - logf(ep + 1e-8f));
        }
        *out_loss = loss;
    }
}

// ======================================================================
// 3) routed expert FFN, fp32 WMMA, M=32 tile (two 16-row tiles share B)
// ======================================================================
__global__ void __launch_bounds__(256)
moe_expert_kernel(const float* __restrict__ xf,       // [T,DM]
                  const float* __restrict__ we1,      // [E,DM,DF]
                  const float* __restrict__ be1,      // [E,DF]
                  const float* __restrict__ we2,      // [E,DF,DM]
                  const float* __restrict__ be2,      // [E,DM]
                  const float* __restrict__ escale,   // [E,DM]
                  const float* __restrict__ ebias,    // [E,DM]
                  const float* __restrict__ gates,    // [T,2]
                  const int*   __restrict__ counts,   // [8]
                  const int*   __restrict__ bucket,   // [8][T]
                  float*       __restrict__ out_slot) // [T,2,DM] gate-weighted
{
    const int e    = blockIdx.y;
    const int cnt  = counts[e];
    const int base = blockIdx.x * MT;
    if (base >= cnt) return;                 // uniform exit, EXEC stays full for WMMA
    const int nv = min(MT, cnt - base);

    extern __shared__ float smemf[];
    float* sx   = smemf;                     // [MT][SXS]
    float* shh  = smemf + MT * SXS;          // [MT][HS]
    int*   sidx = (int*)(shh + MT * HS);     // [MT]  token*2+slot
    float* sg   = (float*)(sidx + MT);       // [MT]  gate weight

    const int tid  = threadIdx.x;
    const int wid  = tid >> 5;               // 0..7
    const int lane = tid & 31;
    const int m    = lane & 15;              // A-row / B-col index within a 16-tile
    const int hi   = lane >> 4;              // half-wave selects K pair {0,1} vs {2,3}
    const int ksel = hi * 2;

    if (tid < MT) {
        if (tid < nv) {
            int si = bucket[(size_t)e * T_TOKENS + base + tid];
            sidx[tid] = si;
            sg[tid]   = gates[si];
        } else {
            sidx[tid] = 0;
            sg[tid]   = 0.0f;
        }
    }
    __syncthreads();

    // ---- async-stage x tile into LDS (16B per lane per op), zero tail rows ----
    for (int i = tid; i < MT * (DM / 4); i += 256) {   // 16B chunks: 256 per row
        const int r = i >> 8;
        const int c = i & 255;
        float* l = sx + r * SXS + c * 4;
        if (r < nv) {
            async_g2l_b128(xf + (size_t)(sidx[r] >> 1) * DM + c * 4, l);
        } else {
            float4 z = make_float4(0.f, 0.f, 0.f, 0.f);
            *(float4*)l = z;
        }
    }
    wait_asynccnt0();
    __syncthreads();

    const float* w1 = we1 + (size_t)e * DM * DF;
    const float* w2 = we2 + (size_t)e * DF * DM;
    const int n0 = wid * 128;                // this wave's 128 output columns

    v8f c2[2][8] = {};                       // 32x128 fp32 accumulators

    for (int ch = 0; ch < DF; ch += FC) {
        // ---- GEMM1: C1[32x16] = X[32x1024] * W1[1024, fbase:fbase+16] ----
        const int fbase = ch + wid * 16;
        const float b1 = be1[(size_t)e * DF + fbase + m];  // hoisted: hides under K-loop
        v8f c1[2] = {};
        const float* w1c = w1 + fbase + m;   // column fbase+m
        #pragma unroll 4
        for (int k0 = 0; k0 < DM; k0 += 4) {
            const int kk = k0 + ksel;
            v2f b;
            const float* wp = w1c + (size_t)kk * DF;
            b.x = wp[0];
            b.y = wp[DF];
            v2f a0, a1;
            a0.x = sx[m * SXS + kk];
            a0.y = sx[m * SXS + kk + 1];
            a1.x = sx[(16 + m) * SXS + kk];
            a1.y = sx[(16 + m) * SXS + kk + 1];
            c1[0] = __builtin_amdgcn_wmma_f32_16x16x4_f32(false, a0, false, b,
                                                          (short)0, c1[0], false, false);
            c1[1] = __builtin_amdgcn_wmma_f32_16x16x4_f32(false, a1, false, b,
                                                          (short)0, c1[1], false, false);
        }
        // bias + branch-free erf-GELU, stage h-chunk to LDS
        #pragma unroll
        for (int rt = 0; rt < 2; ++rt) {
            #pragma unroll
            for (int r = 0; r < 8; ++r) {
                float v = gelu_erf(c1[rt][r] + b1);
                shh[(rt * 16 + r + hi * 8) * HS + wid * 16 + m] = v;
            }
        }
        __syncthreads();

        // ---- GEMM2 partial: C2[32x128] += h[32xFC] * W2[ch:ch+FC, n0:n0+128] ----
        #pragma unroll 2
        for (int kc = 0; kc < FC; kc += 4) {
            const int kk = kc + ksel;
            v2f a0, a1;
            a0.x = shh[m * HS + kk];
            a0.y = shh[m * HS + kk + 1];
            a1.x = shh[(16 + m) * HS + kk];
            a1.y = shh[(16 + m) * HS + kk + 1];
            const float* wp2 = w2 + (size_t)(ch + kk) * DM + n0 + m;
            #pragma unroll
            for (int j = 0; j < 8; ++j) {
                v2f b;
                b.x = wp2[j * 16];
                b.y = wp2[j * 16 + DM];
                c2[0][j] = __builtin_amdgcn_wmma_f32_16x16x4_f32(false, a0, false, b,
                                                                (short)0, c2[0][j], false, false);
                c2[1][j] = __builtin_amdgcn_wmma_f32_16x16x4_f32(false, a1, false, b,
                                                                (short)0, c2[1][j], false, false);
            }
        }
        __syncthreads();                     // protect shh before next chunk rewrites it
    }

    // ---- epilogue: +be2, *escale, +ebias, *gate, scatter to (token,slot) ----
    #pragma unroll
    for (int j = 0; j < 8; ++j) {
        const int d = n0 + j * 16 + m;
        const float b2 = be2[(size_t)e * DM + d];
        const float es = escale[(size_t)e * DM + d];
        const float eb = ebias[(size_t)e * DM + d];
        #pragma unroll
        for (int rt = 0; rt < 2; ++rt) {
            #pragma unroll
            for (int r = 0; r < 8; ++r) {
                const int row = rt * 16 + r + hi * 8;
                if (row < nv) {
                    float v = (c2[rt][j][r] + b2) * es + eb;
                    out_slot[(size_t)sidx[row] * DM + d] = sg[row] * v;
                }
            }
        }
    }
}

// ======================================================================
// 4) residual + combine + LayerNorm; one block per token
// ======================================================================
__global__ void moe_ln_kernel(const float* __restrict__ x,
                              const float* __restrict__ out_slot, // [T,2,DM]
                              const float* __restrict__ gamma,
                              const float* __restrict__ beta,
                              float* __restrict__ y)
{
    const int t   = blockIdx.x;
    const int tid = threadIdx.x;
    __shared__ float red[256];
    __shared__ float bcast;

    float v[4];
    #pragma unroll
    for (int ii = 0; ii < 4; ++ii) {
        const int d = ii * 256 + tid;
        v[ii] = x[(size_t)t * DM + d]
              + out_slot[((size_t)t * 2 + 0) * DM + d]
              + out_slot[((size_t)t * 2 + 1) * DM + d];
    }
    float local = v[0] + v[1] + v[2] + v[3];
    red[tid] = local; __syncthreads();
    for (int s = 128; s > 0; s >>= 1) {
        if (tid < s) red[tid] += red[tid + s];
        __syncthreads();
    }
    if (tid == 0) bcast = red[0] * (1.0f / (float)DM);
    __syncthreads();
    const float mu = bcast;
    __syncthreads();

    float ls = 0.0f;
    #pragma unroll
    for (int ii = 0; ii < 4; ++ii) { float dv = v[ii] - mu; ls += dv * dv; }
    red[tid] = ls; __syncthreads();
    for (int s = 128; s > 0; s >>= 1) {
        if (tid < s) red[tid] += red[tid + s];
        __syncthreads();
    }
    if (tid == 0) bcast = rsqrtf(red[0] * (1.0f / (float)DM) + LN_EPS);
    __syncthreads();
    const float rs = bcast;

    #pragma unroll
    for (int ii = 0; ii < 4; ++ii) {
        const int d = ii * 256 + tid;
        y[(size_t)t * DM + d] = (v[ii] - mu) * rs * gamma[d] + beta[d];
    }
}

// ======================================================================
// host launcher
// ======================================================================
extern "C" void kernel_launch(void* const* d_in, const int* in_sizes, int n_in,
                              void* d_out, int out_size, void* d_ws, size_t ws_size,
                              hipStream_t stream) {
    const float* x      = (const float*)d_in[0];
    const float* gw1    = (const float*)d_in[1];
    const float* gb1    = (const float*)d_in[2];
    const float* gw2    = (const float*)d_in[3];
    const float* gb2    = (const float*)d_in[4];
    const float* we1    = (const float*)d_in[5];
    const float* be1    = (const float*)d_in[6];
    const float* we2    = (const float*)d_in[7];
    const float* be2    = (const float*)d_in[8];
    const float* escale = (const float*)d_in[9];
    const float* ebias  = (const float*)d_in[10];
    const float* gamma  = (const float*)d_in[11];
    const float* beta   = (const float*)d_in[12];
    float* y = (float*)d_out;
    float* loss = y + (size_t)T_TOKENS * DM;

    // workspace layout
    char* w = (char*)d_ws;
    float* probs = (float*)w;     w += (size_t)T_TOKENS * NUM_E * 4;
    float* gates = (float*)w;     w += (size_t)T_TOKENS * 2 * 4;
    int*   counts = (int*)w;      w += 256;
    int*   bucket = (int*)w;      w += (size_t)NUM_E * T_TOKENS * 4;
    float* out_slot = (float*)w;  w += (size_t)T_TOKENS * 2 * DM * 4;

    (void)hipFuncSetAttribute((const void*)moe_expert_kernel,
                              hipFuncAttributeMaxDynamicSharedMemorySize,
                              EXPERT_SMEM_BYTES);

    moe_zero_kernel<<<1, 64, 0, stream>>>(counts);
    moe_gating_kernel<<<T_TOKENS, 256, 0, stream>>>(x, gw1, gb1, gw2, gb2,
                                                    probs, gates, counts, bucket);
    moe_loss_kernel<<<1, 256, 0, stream>>>(probs, loss);
    dim3 egrid(T_TOKENS / MT, NUM_E, 1);   // worst case: all tokens on one expert
    moe_expert_kernel<<<egrid, 256, EXPERT_SMEM_BYTES, stream>>>(
        x, we1, be1, we2, be2, escale, ebias, gates, counts, bucket, out_slot);
    moe_ln_kernel<<<T_TOKENS, 256, 0, stream>>>(x, out_slot, gamma, beta, y);
}